// BatchedLIDIA_38972533244524
// MI455X (gfx1250) — compile-verified
//
#include <hip/hip_runtime.h>
#include <math.h>

// ---------------- problem constants (fixed shapes from reference) ----------
#define T_  4
#define C_  3
#define H_  128
#define W_  128
#define PS_ 5
#define K_  14
#define WS_ 15
#define SW_ 7      // WS/2
#define PW_ 2      // PS/2
#define RAD_ 9     // SW+PW
#define HP_ 146    // H + 2*RAD
#define HPWP_ (HP_*HP_)   // 21316
#define NOFF_ (WS_*WS_)   // 225

typedef float v2f __attribute__((ext_vector_type(2)));
typedef float v8f __attribute__((ext_vector_type(8)));

// Exact fp32 WMMA: D(16x16) = A(16x4) * B(4x16) + C   (V_WMMA_F32_16X16X4_F32)
__device__ __forceinline__ v8f wmma_f32_16x16x4(v2f a, v2f b, v8f c) {
#if defined(__has_builtin)
#if __has_builtin(__builtin_amdgcn_wmma_f32_16x16x4_f32)
  return __builtin_amdgcn_wmma_f32_16x16x4_f32(false, a, false, b, (short)0, c,
                                               false, false);
#else
  asm volatile("v_wmma_f32_16x16x4_f32 %0, %1, %2, %0"
               : "+v"(c) : "v"(a), "v"(b));
  return c;
#endif
#else
  asm volatile("v_wmma_f32_16x16x4_f32 %0, %1, %2, %0"
               : "+v"(c) : "v"(a), "v"(b));
  return c;
#endif
}

// ---------------- kernel 1: per-(t,c) mean of normalized image -------------
__global__ void lidia_means(const float* __restrict__ noisy,
                            float* __restrict__ means) {
  __shared__ float red[256];
  const int tc = blockIdx.x;           // 0..11
  const float* src = noisy + (size_t)tc * (H_ * W_);
  float s = 0.f;
  for (int i = threadIdx.x; i < H_ * W_; i += 256) s += src[i];
  red[threadIdx.x] = s;
  __syncthreads();
  for (int off = 128; off > 0; off >>= 1) {
    if (threadIdx.x < off) red[threadIdx.x] += red[threadIdx.x + off];
    __syncthreads();
  }
  if (threadIdx.x == 0) {
    // mean of x = mean(noisy)/127.5 - 1
    means[tc] = red[0] / (float)(H_ * W_) / 127.5f - 1.f;
  }
}

// ---------------- kernel 2: build reflect-padded, mean-subtracted P --------
__device__ __forceinline__ int reflect128(int i) {
  if (i < 0) return -i;
  if (i > H_ - 1) return 2 * (H_ - 1) - i;
  return i;
}

__global__ void lidia_build_p(const float* __restrict__ noisy,
                              const float* __restrict__ means,
                              float* __restrict__ P) {
  int gid = blockIdx.x * blockDim.x + threadIdx.x;
  const int total = T_ * C_ * HPWP_;
  if (gid >= total) return;
  int tc  = gid / HPWP_;
  int rem = gid - tc * HPWP_;
  int Y = rem / HP_;
  int X = rem - Y * HP_;
  int yy = reflect128(Y - RAD_);
  int xx = reflect128(X - RAD_);
  float v = noisy[(size_t)tc * (H_ * W_) + yy * W_ + xx] / 127.5f - 1.f;
  P[gid] = v - means[tc];
}

// ---------------- kernel 3: fused search (E -> WMMA box filter -> top-k) ---
// one wave (32 threads) per 16x16 output tile per t
__global__ __launch_bounds__(32)
void lidia_search(const float* __restrict__ P,
                  const float* __restrict__ sigma,
                  float* __restrict__ wts,
                  int* __restrict__ oidx) {
  __shared__ float Pl[C_][36][36];      // 35x35 subtile of P, padded stride
  __shared__ float Ebuf[32 * 20];       // E tile, rows 20..31 stay zero
  __shared__ float Tbuf[32 * 16];       // Tmp = E * A5^T
  __shared__ float dlist[256 * K_];     // per-pixel sorted distances
  __shared__ int   olist[256 * K_];     // per-pixel offsets

  const int lane = threadIdx.x & 31;
  const int m16  = lane & 15;
  const int half = lane >> 4;           // 0 or 1
  const int tile = blockIdx.x;          // 0..63
  const int t    = blockIdx.y;          // 0..3
  const int y0   = (tile >> 3) * 16;    // tile origin in D coords
  const int x0   = (tile & 7) * 16;

  // ---- load 35x35x3 subtile of P; P rows [y0, y0+35), cols [x0, x0+35)
  const float* Pt = P + (size_t)t * C_ * HPWP_;
  for (int ii = lane; ii < C_ * 35 * 35; ii += 32) {
    int c  = ii / (35 * 35);
    int r  = (ii - c * 35 * 35) / 35;
    int cc = ii - c * 35 * 35 - r * 35;
    Pl[c][r][cc] = Pt[(size_t)c * HPWP_ + (y0 + r) * HP_ + (x0 + cc)];
  }
  // ---- init top-k lists and zero E rows (incl. pad rows 20..31)
  for (int ii = lane; ii < 256 * K_; ii += 32) {
    dlist[ii] = 3.0e38f;
    olist[ii] = 0;
  }
  for (int ii = lane; ii < 32 * 20; ii += 32) Ebuf[ii] = 0.f;
  __syncthreads();

  // ---- offset loop ----
  for (int o = 0; o < NOFF_; ++o) {
    const int oy = o / WS_;
    const int ox = o - oy * WS_;

    __syncthreads();  // E writes vs previous-iteration reads
    // E(ey,ex) = sum_c (P[SW+y][SW+x] - P[oy+y][ox+x])^2 , 20x20
    for (int ii = lane; ii < 20 * 20; ii += 32) {
      int ey = ii / 20;
      int ex = ii - ey * 20;
      float s = 0.f;
#pragma unroll
      for (int c = 0; c < C_; ++c) {
        float d = Pl[c][ey + SW_][ex + SW_] - Pl[c][ey + oy][ex + ox];
        s += d * d;
      }
      Ebuf[ey * 20 + ex] = s;
    }
    __syncthreads();

    // ---- GEMM1: Tmp(20x16) = E(20x20) * A5^T(20x16), two M-tiles of 16 ----
    // A-frag element v: (row = Mtile*16 + m16, K = 4*kk + 2*half + v) from Ebuf
    // B-frag element v: B[k][n] = A5[n][k] = ((k-n) in [0,5))
#pragma unroll
    for (int Mtile = 0; Mtile < 2; ++Mtile) {
      v8f acc = {};
#pragma unroll
      for (int kk = 0; kk < 5; ++kk) {
        const int kb = 4 * kk + 2 * half;
        v2f a, b;
        a.x = Ebuf[(Mtile * 16 + m16) * 20 + kb + 0];
        a.y = Ebuf[(Mtile * 16 + m16) * 20 + kb + 1];
        int d0 = kb + 0 - m16;
        int d1 = kb + 1 - m16;
        b.x = (d0 >= 0 && d0 < PS_) ? 1.f : 0.f;
        b.y = (d1 >= 0 && d1 < PS_) ? 1.f : 0.f;
        acc = wmma_f32_16x16x4(a, b, acc);
      }
#pragma unroll
      for (int v = 0; v < 8; ++v)
        Tbuf[(Mtile * 16 + v + 8 * half) * 16 + m16] = acc[v];
    }
    __syncthreads();

    // ---- GEMM2: Dtile(16x16) = A5(16x20) * Tmp(20x16) ----
    v8f dacc = {};
#pragma unroll
    for (int kk = 0; kk < 5; ++kk) {
      const int kb = 4 * kk + 2 * half;
      v2f a, b;
      int d0 = kb + 0 - m16;
      int d1 = kb + 1 - m16;
      a.x = (d0 >= 0 && d0 < PS_) ? 1.f : 0.f;
      a.y = (d1 >= 0 && d1 < PS_) ? 1.f : 0.f;
      b.x = Tbuf[(kb + 0) * 16 + m16];
      b.y = Tbuf[(kb + 1) * 16 + m16];
      dacc = wmma_f32_16x16x4(a, b, dacc);
    }

    // ---- top-k insert: lane owns pixels (row = v + 8*half, col = m16) ----
#pragma unroll
    for (int v = 0; v < 8; ++v) {
      const int pix = (v + 8 * half) * 16 + m16;
      const float d = dacc[v];
      if (d < dlist[pix * K_ + (K_ - 1)]) {
        int kp = K_ - 1;
        while (kp > 0 && dlist[pix * K_ + kp - 1] > d) {
          dlist[pix * K_ + kp] = dlist[pix * K_ + kp - 1];
          olist[pix * K_ + kp] = olist[pix * K_ + kp - 1];
          --kp;
        }
        dlist[pix * K_ + kp] = d;
        olist[pix * K_ + kp] = o;
      }
    }
  }
  __syncthreads();

  // ---- softmax weights + store (weights already normalized) ----
  const float sig = sigma[0] / 255.f / 0.5f;
  const float denom = 2.f * (float)(C_ * PS_ * PS_) * sig * sig + 1e-08f;
#pragma unroll
  for (int v = 0; v < 8; ++v) {
    const int py = v + 8 * half;
    const int pix = py * 16 + m16;
    const float d0 = dlist[pix * K_ + 0];
    float e[K_];
    float s = 0.f;
#pragma unroll
    for (int k = 0; k < K_; ++k) {
      e[k] = expf(-(dlist[pix * K_ + k] - d0) / denom);
      s += e[k];
    }
    const float inv = 1.f / s;
    const int gy = y0 + py;
    const int gx = x0 + m16;
    const size_t b = (((size_t)t * H_ + gy) * W_ + gx) * K_;
#pragma unroll
    for (int k = 0; k < K_; ++k) {
      wts[b + k]  = e[k] * inv;
      oidx[b + k] = olist[pix * K_ + k];
    }
  }
}

// ---------------- kernel 4: gather reconstruction + denormalize ------------
__global__ void lidia_reconstruct(const float* __restrict__ P,
                                  const float* __restrict__ means,
                                  const float* __restrict__ wts,
                                  const int* __restrict__ oidx,
                                  float* __restrict__ out) {
  int g = blockIdx.x * blockDim.x + threadIdx.x;
  if (g >= T_ * H_ * W_) return;
  const int t = g / (H_ * W_);
  const int rem = g - t * (H_ * W_);
  const int y = rem / W_;
  const int x = rem - y * W_;
  const int Yp = y + PW_;  // coords in the (132x132) accumulation space
  const int Xp = x + PW_;

  float acc0 = 0.f, acc1 = 0.f, acc2 = 0.f, cnt = 0.f;
  const float* Pt0 = P + (size_t)(t * C_ + 0) * HPWP_;
  const float* Pt1 = P + (size_t)(t * C_ + 1) * HPWP_;
  const float* Pt2 = P + (size_t)(t * C_ + 2) * HPWP_;

#pragma unroll
  for (int py = 0; py < PS_; ++py) {
    const int i = Yp - py;
    if ((unsigned)i >= (unsigned)H_) continue;
#pragma unroll
    for (int px = 0; px < PS_; ++px) {
      const int j = Xp - px;
      if ((unsigned)j >= (unsigned)W_) continue;
      cnt += 1.f;
      const size_t b = (((size_t)t * H_ + i) * W_ + j) * K_;
#pragma unroll
      for (int k = 0; k < K_; ++k) {
        const float w = wts[b + k];
        const int o = oidx[b + k];
        const int oy = o / WS_;
        const int ox = o - oy * WS_;
        // patch read row collapses to Yp+oy (independent of py): pr = i+oy+py
        const int pr = Yp + oy;
        const int pc = Xp + ox;
        const int a = pr * HP_ + pc;
        acc0 += w * Pt0[a];
        acc1 += w * Pt1[a];
        acc2 += w * Pt2[a];
      }
    }
  }
  const float invc = 1.f / (cnt + 1e-10f);
  const float d0 = acc0 * invc + means[t * C_ + 0];
  const float d1 = acc1 * invc + means[t * C_ + 1];
  const float d2 = acc2 * invc + means[t * C_ + 2];
  float* ot = out + (size_t)t * C_ * H_ * W_;
  ot[(size_t)0 * H_ * W_ + y * W_ + x] = 255.f * (d0 * 0.5f + 0.5f);
  ot[(size_t)1 * H_ * W_ + y * W_ + x] = 255.f * (d1 * 0.5f + 0.5f);
  ot[(size_t)2 * H_ * W_ + y * W_ + x] = 255.f * (d2 * 0.5f + 0.5f);
}

// ---------------- launcher -------------------------------------------------
extern "C" void kernel_launch(void* const* d_in, const int* in_sizes, int n_in,
                              void* d_out, int out_size, void* d_ws,
                              size_t ws_size, hipStream_t stream) {
  const float* noisy = (const float*)d_in[0];  // (4,3,128,128) f32
  const float* sigma = (const float*)d_in[1];  // (1,) f32
  float* out = (float*)d_out;                  // (4,3,128,128) f32

  // workspace layout (all 64B aligned):
  //   means : 12 f32                      @ 0
  //   P     : 4*3*146*146 f32 (1.02 MB)   @ 1024
  //   wts   : 4*128*128*14 f32 (3.67 MB)  @ OFF_WTS
  //   oidx  : 4*128*128*14 i32 (3.67 MB)  @ OFF_IDX
  char* ws = (char*)d_ws;
  const size_t P_BYTES = (size_t)T_ * C_ * HPWP_ * sizeof(float); // 1,023,168
  const size_t W_BYTES = (size_t)T_ * H_ * W_ * K_ * sizeof(float); // 3,670,016
  const size_t OFF_P   = 1024;
  const size_t OFF_WTS = (OFF_P + P_BYTES + 63) & ~(size_t)63;
  const size_t OFF_IDX = (OFF_WTS + W_BYTES + 63) & ~(size_t)63;
  float* means = (float*)(ws);
  float* P     = (float*)(ws + OFF_P);
  float* wts   = (float*)(ws + OFF_WTS);
  int*   oidx  = (int*)(ws + OFF_IDX);

  lidia_means<<<T_ * C_, 256, 0, stream>>>(noisy, means);

  const int pTotal = T_ * C_ * HPWP_;
  lidia_build_p<<<(pTotal + 255) / 256, 256, 0, stream>>>(noisy, means, P);

  dim3 sgrid((H_ / 16) * (W_ / 16), T_);   // 64 tiles x 4 frames
  lidia_search<<<sgrid, 32, 0, stream>>>(P, sigma, wts, oidx);

  lidia_reconstruct<<<(T_ * H_ * W_ + 255) / 256, 256, 0, stream>>>(
      P, means, wts, oidx, out);
}